// GraphModel_65781719105989
// MI455X (gfx1250) — compile-verified
//
#include <hip/hip_runtime.h>

#define N_NODES 100000
#define N_EDGES 3200000
#define R_REL   9
#define G_DIM   768
#define H1_DIM  64
#define H2_DIM  32
#define C_DIM   4

typedef __attribute__((ext_vector_type(2))) float v2f;
typedef __attribute__((ext_vector_type(8))) float v8f;

// D = A(16x4, f32) * B(4x16, f32) + C(16x16, f32)
__device__ __forceinline__ v8f wmma_f32_step(v2f a, v2f b, v8f c) {
  return __builtin_amdgcn_wmma_f32_16x16x4_f32(
      /*neg_a=*/false, a, /*neg_b=*/false, b,
      /*c_mod=*/(short)0, c, /*reuse_a=*/false, /*reuse_b=*/false);
}

// ---------------------------------------------------------------------------
// fill
// ---------------------------------------------------------------------------
__global__ void fill_kernel(float* __restrict__ p, float v, int n) {
  int i = blockIdx.x * blockDim.x + threadIdx.x;
  if (i < n) p[i] = v;
}

// ---------------------------------------------------------------------------
// h1 = x @ Wp + bp   [N,768]x[768,64]
// one wave -> one 16x64 output row-slab: A fragment loaded ONCE per k-step and
// reused across the 4 column-tile WMMAs -> x streamed from HBM exactly once.
// ---------------------------------------------------------------------------
__global__ void proj_kernel(const float* __restrict__ x,
                            const float* __restrict__ Wp,
                            const float* __restrict__ bp,
                            float* __restrict__ h1) {
  const int wid  = (blockIdx.x * blockDim.x + threadIdx.x) >> 5;  // row tile
  const int lane = threadIdx.x & 31;
  if (wid >= N_NODES / 16) return;

  const int row  = wid * 16 + (lane & 15);
  const int kh   = lane >> 4;        // 0/1 -> K pair selector
  const int colL = lane & 15;

  const float* xr = x  + (size_t)row * G_DIM + kh * 2;
  const float* wb = Wp + (size_t)(kh * 2) * H1_DIM + colL;

  v8f acc[4];
#pragma unroll
  for (int t = 0; t < 4; ++t) acc[t] = (v8f){0.f,0.f,0.f,0.f,0.f,0.f,0.f,0.f};

#pragma unroll 2
  for (int k = 0; k < G_DIM; k += 4) {
    v2f a; a.x = xr[k]; a.y = xr[k + 1];
    const float* w0 = wb + (size_t)k * H1_DIM;  // row k+kh*2
    const float* w1 = w0 + H1_DIM;              // row k+kh*2+1
#pragma unroll
    for (int t = 0; t < 4; ++t) {
      v2f b; b.x = w0[t * 16]; b.y = w1[t * 16];
      acc[t] = wmma_f32_step(a, b, acc[t]);
    }
  }

  const int rbase = wid * 16 + kh * 8;
#pragma unroll
  for (int t = 0; t < 4; ++t) {
    const int col = t * 16 + colL;
    const float bias = bp[col];
#pragma unroll
    for (int r = 0; r < 8; ++r)
      h1[(size_t)(rbase + r) * H1_DIM + col] = acc[t][r] + bias;
  }
}

// ---------------------------------------------------------------------------
// per-(relation,dst) in-degree counts
// ---------------------------------------------------------------------------
__global__ void count_kernel(const int* __restrict__ ei,
                             const int* __restrict__ et,
                             float* __restrict__ cnt) {
  int e = blockIdx.x * blockDim.x + threadIdx.x;
  if (e >= N_EDGES) return;
  int dst = ei[N_EDGES + e];
  int r   = et[e];
  atomicAdd(&cnt[(size_t)r * N_NODES + dst], 1.0f);
}

// ---------------------------------------------------------------------------
// RGCN messages: agg[dst] += (h1[src] @ W_rel[r]) / deg_r(dst)
// one wave per edge; lane = output channel (H2=32); h1[src][k] wave-uniform
// ---------------------------------------------------------------------------
__global__ void rgcn_msg_kernel(const int* __restrict__ ei,
                                const int* __restrict__ et,
                                const float* __restrict__ h1,
                                const float* __restrict__ Wrel,
                                const float* __restrict__ cnt,
                                float* __restrict__ agg) {
  const int e    = (blockIdx.x * blockDim.x + threadIdx.x) >> 5;
  const int lane = threadIdx.x & 31;
  if (e >= N_EDGES) return;
  const int src = ei[e];
  const int dst = ei[N_EDGES + e];
  const int r   = et[e];
  const float norm = 1.0f / fmaxf(cnt[(size_t)r * N_NODES + dst], 1.0f);

  const float* hs = h1 + (size_t)src * H1_DIM;                 // wave-uniform
  const float* W  = Wrel + (size_t)r * H1_DIM * H2_DIM + lane; // per-lane col

  float acc = 0.f;
#pragma unroll
  for (int kk = 0; kk < H1_DIM; ++kk)
    acc = fmaf(hs[kk], W[kk * H2_DIM], acc);

  atomicAdd(&agg[(size_t)dst * H2_DIM + lane], acc * norm);
}

// ---------------------------------------------------------------------------
// h2 = relu(agg + h1 @ W_root + b)   [N,64]x[64,32] via WMMA f32
// one wave -> 16x32 slab (both column tiles), A fragment reused
// ---------------------------------------------------------------------------
__global__ void rgcn_combine_kernel(const float* __restrict__ h1,
                                    const float* __restrict__ Wroot,
                                    const float* __restrict__ b,
                                    const float* __restrict__ agg,
                                    float* __restrict__ h2) {
  const int wid  = (blockIdx.x * blockDim.x + threadIdx.x) >> 5;
  const int lane = threadIdx.x & 31;
  if (wid >= N_NODES / 16) return;

  const int row  = wid * 16 + (lane & 15);
  const int kh   = lane >> 4;
  const int colL = lane & 15;

  const float* ar = h1    + (size_t)row * H1_DIM + kh * 2;
  const float* wb = Wroot + (size_t)(kh * 2) * H2_DIM + colL;

  v8f acc[2];
  acc[0] = (v8f){0.f,0.f,0.f,0.f,0.f,0.f,0.f,0.f};
  acc[1] = acc[0];

#pragma unroll
  for (int k = 0; k < H1_DIM; k += 4) {
    v2f a; a.x = ar[k]; a.y = ar[k + 1];
    const float* w0 = wb + (size_t)k * H2_DIM;
    const float* w1 = w0 + H2_DIM;
#pragma unroll
    for (int t = 0; t < 2; ++t) {
      v2f bb; bb.x = w0[t * 16]; bb.y = w1[t * 16];
      acc[t] = wmma_f32_step(a, bb, acc[t]);
    }
  }

  const int rbase = wid * 16 + kh * 8;
#pragma unroll
  for (int t = 0; t < 2; ++t) {
    const int col = t * 16 + colL;
    const float bias = b[col];
#pragma unroll
    for (int r = 0; r < 8; ++r) {
      size_t idx = (size_t)(rbase + r) * H2_DIM + col;
      h2[idx] = fmaxf(acc[t][r] + agg[idx] + bias, 0.f);
    }
  }
}

// ---------------------------------------------------------------------------
// fused q/k/v/skip projections: O_m = h2 @ W_m + b_m  (4x [N,32]x[32,32])
// one wave -> 16-row slab of all four outputs; h2 read once
// ---------------------------------------------------------------------------
__global__ void qkvs_kernel(const float* __restrict__ h2,
                            const float* __restrict__ Wq, const float* __restrict__ bq,
                            const float* __restrict__ Wk, const float* __restrict__ bk,
                            const float* __restrict__ Wv, const float* __restrict__ bv,
                            const float* __restrict__ Ws, const float* __restrict__ bs,
                            float* __restrict__ q, float* __restrict__ k,
                            float* __restrict__ v, float* __restrict__ s) {
  const int wid  = (blockIdx.x * blockDim.x + threadIdx.x) >> 5;
  const int lane = threadIdx.x & 31;
  if (wid >= N_NODES / 16) return;

  const int row  = wid * 16 + (lane & 15);
  const int kh   = lane >> 4;
  const int colL = lane & 15;

  const float* W[4] = {Wq, Wk, Wv, Ws};
  const float* B[4] = {bq, bk, bv, bs};
  float*       O[4] = {q, k, v, s};

  const float* ar = h2 + (size_t)row * H2_DIM + kh * 2;
  const size_t wofs = (size_t)(kh * 2) * H2_DIM + colL;

  v8f acc[4][2];
#pragma unroll
  for (int m = 0; m < 4; ++m)
#pragma unroll
    for (int t = 0; t < 2; ++t)
      acc[m][t] = (v8f){0.f,0.f,0.f,0.f,0.f,0.f,0.f,0.f};

#pragma unroll
  for (int kk = 0; kk < H2_DIM; kk += 4) {
    v2f a; a.x = ar[kk]; a.y = ar[kk + 1];
#pragma unroll
    for (int m = 0; m < 4; ++m) {
      const float* w0 = W[m] + wofs + (size_t)kk * H2_DIM;
      const float* w1 = w0 + H2_DIM;
#pragma unroll
      for (int t = 0; t < 2; ++t) {
        v2f bb; bb.x = w0[t * 16]; bb.y = w1[t * 16];
        acc[m][t] = wmma_f32_step(a, bb, acc[m][t]);
      }
    }
  }

  const int rbase = wid * 16 + kh * 8;
#pragma unroll
  for (int m = 0; m < 4; ++m)
#pragma unroll
    for (int t = 0; t < 2; ++t) {
      const int col = t * 16 + colL;
      const float bias = B[m][col];
#pragma unroll
      for (int r = 0; r < 8; ++r)
        O[m][(size_t)(rbase + r) * H2_DIM + col] = acc[m][t][r] + bias;
    }
}

// ---------------------------------------------------------------------------
// logits[e] = dot(q[dst], k[src]) / sqrt(32); segment-max via encoded-int atomicMax
// one wave per edge; lane = channel; wave reduction via shfl_xor (wave32)
// ---------------------------------------------------------------------------
__global__ void logits_max_kernel(const int* __restrict__ ei,
                                  const float* __restrict__ q,
                                  const float* __restrict__ k,
                                  float* __restrict__ logits,
                                  int* __restrict__ mEnc) {
  const int e    = (blockIdx.x * blockDim.x + threadIdx.x) >> 5;
  const int lane = threadIdx.x & 31;
  if (e >= N_EDGES) return;
  const int src = ei[e];
  const int dst = ei[N_EDGES + e];

  float xv = q[(size_t)dst * H2_DIM + lane] * k[(size_t)src * H2_DIM + lane];
#pragma unroll
  for (int off = 16; off > 0; off >>= 1)
    xv += __shfl_xor(xv, off, 32);
  xv *= 0.17677669529663687f;   // 1/sqrt(32)

  if (lane == 0) {
    logits[e] = xv;
    int enc = __float_as_int(xv);
    enc = enc >= 0 ? enc : (enc ^ 0x7fffffff);   // order-preserving encoding
    atomicMax(&mEnc[dst], enc);
  }
}

// ---------------------------------------------------------------------------
// p[e] = exp(logit - m[dst]);  denom[dst] += p[e]   (in-place over logits)
// ---------------------------------------------------------------------------
__global__ void exp_denom_kernel(const int* __restrict__ ei,
                                 const int* __restrict__ mEnc,
                                 float* __restrict__ lp,
                                 float* __restrict__ denom) {
  int e = blockIdx.x * blockDim.x + threadIdx.x;
  if (e >= N_EDGES) return;
  int dst = ei[N_EDGES + e];
  int enc = mEnc[dst];
  float mv = __int_as_float(enc >= 0 ? enc : (enc ^ 0x7fffffff));
  float pe = __expf(lp[e] - mv);
  lp[e] = pe;
  atomicAdd(&denom[dst], pe);
}

// ---------------------------------------------------------------------------
// attn[dst] += (p[e]/denom[dst]) * v[src]   one wave per edge, lane = channel
// ---------------------------------------------------------------------------
__global__ void attn_scatter_kernel(const int* __restrict__ ei,
                                    const float* __restrict__ p,
                                    const float* __restrict__ denom,
                                    const float* __restrict__ v,
                                    float* __restrict__ attn) {
  const int e    = (blockIdx.x * blockDim.x + threadIdx.x) >> 5;
  const int lane = threadIdx.x & 31;
  if (e >= N_EDGES) return;
  const int src = ei[e];
  const int dst = ei[N_EDGES + e];
  const float alpha = p[e] / denom[dst];
  atomicAdd(&attn[(size_t)dst * H2_DIM + lane],
            alpha * v[(size_t)src * H2_DIM + lane]);
}

// ---------------------------------------------------------------------------
// out = relu(attn + skip_pre) @ Wf + bf   (H2=32 -> C=4), one thread per node
// ---------------------------------------------------------------------------
__global__ void final_kernel(const float* __restrict__ attn,
                             const float* __restrict__ skip,
                             const float* __restrict__ Wf,
                             const float* __restrict__ bf,
                             float* __restrict__ out) {
  int n = blockIdx.x * blockDim.x + threadIdx.x;
  if (n >= N_NODES) return;
  float hb[H2_DIM];
#pragma unroll
  for (int j = 0; j < H2_DIM; ++j)
    hb[j] = fmaxf(attn[(size_t)n * H2_DIM + j] + skip[(size_t)n * H2_DIM + j], 0.f);
#pragma unroll
  for (int c = 0; c < C_DIM; ++c) {
    float acc = bf[c];
#pragma unroll
    for (int j = 0; j < H2_DIM; ++j)
      acc = fmaf(hb[j], Wf[j * C_DIM + c], acc);
    out[(size_t)n * C_DIM + c] = acc;
  }
}

// ---------------------------------------------------------------------------
extern "C" void kernel_launch(void* const* d_in, const int* in_sizes, int n_in,
                              void* d_out, int out_size, void* d_ws, size_t ws_size,
                              hipStream_t stream) {
  const float* x     = (const float*)d_in[0];
  // d_in[1] = lengths: unused by reference
  const int*   ei    = (const int*)d_in[2];
  const int*   et    = (const int*)d_in[3];
  const float* Wp    = (const float*)d_in[4];
  const float* bp    = (const float*)d_in[5];
  const float* Wrel  = (const float*)d_in[6];
  const float* Wroot = (const float*)d_in[7];
  const float* brg   = (const float*)d_in[8];
  const float* Wq    = (const float*)d_in[9];
  const float* bq    = (const float*)d_in[10];
  const float* Wk    = (const float*)d_in[11];
  const float* bk    = (const float*)d_in[12];
  const float* Wv    = (const float*)d_in[13];
  const float* bv    = (const float*)d_in[14];
  const float* Wsk   = (const float*)d_in[15];
  const float* bsk   = (const float*)d_in[16];
  const float* Wf    = (const float*)d_in[17];
  const float* bfb   = (const float*)d_in[18];
  float* out = (float*)d_out;

  // workspace layout (floats); regions reused once their producers/consumers retire
  float* ws   = (float*)d_ws;
  float* h1   = ws;                    // N*64 = 6,400,000
  float* h2   = h1  + 6400000;         // N*32 = 3,200,000
  float* cnt  = h2  + 3200000;         // R*N  =   900,000
  float* agg  = cnt + 900000;          // N*32 -> reused as logits/p (E floats)
  float* vbuf = agg + 3200000;         // N*32
  float* skip = vbuf + 3200000;        // N*32
  float* mbuf = skip + 3200000;        // N encoded-int maxes
  float* den  = mbuf + 100000;         // N
  float* qbuf = h1;                    // reuse: h1 dead after combine
  float* kbuf = h1 + 3200000;
  float* attn = h1;                    // reuse: q dead after logits pass
  float* logitsp = agg;                // reuse: agg dead after combine
  int*   mEnc = (int*)mbuf;

  const int TPB = 256;
  const int edge_wave_blocks = (N_EDGES * 32) / TPB;                  // 400000
  const int edge_thr_blocks  = N_EDGES / TPB;                         // 12500
  const int slab_blocks      = ((N_NODES / 16) * 32 + TPB - 1) / TPB; // 782

  // init accumulators
  fill_kernel<<<(900000 + TPB - 1) / TPB, TPB, 0, stream>>>(cnt, 0.f, 900000);
  fill_kernel<<<(3200000 + TPB - 1) / TPB, TPB, 0, stream>>>(agg, 0.f, 3200000);
  fill_kernel<<<(100000 + TPB - 1) / TPB, TPB, 0, stream>>>(mbuf, -0.0f, 100000); // INT_MIN bits
  fill_kernel<<<(100000 + TPB - 1) / TPB, TPB, 0, stream>>>(den, 0.f, 100000);

  // 1) feature projection (WMMA f32, x streamed once)
  proj_kernel<<<slab_blocks, TPB, 0, stream>>>(x, Wp, bp, h1);

  // 2) RGCN: degree counts, messages, combine
  count_kernel<<<edge_thr_blocks, TPB, 0, stream>>>(ei, et, cnt);
  rgcn_msg_kernel<<<edge_wave_blocks, TPB, 0, stream>>>(ei, et, h1, Wrel, cnt, agg);
  rgcn_combine_kernel<<<slab_blocks, TPB, 0, stream>>>(h1, Wroot, brg, agg, h2);

  // 3) fused q/k/v/skip projections (WMMA f32, h2 read once)
  qkvs_kernel<<<slab_blocks, TPB, 0, stream>>>(h2, Wq, bq, Wk, bk, Wv, bv, Wsk, bsk,
                                               qbuf, kbuf, vbuf, skip);

  // 4) attention: logits+max, exp+denom, weighted scatter
  logits_max_kernel<<<edge_wave_blocks, TPB, 0, stream>>>(ei, qbuf, kbuf, logitsp, mEnc);
  fill_kernel<<<(3200000 + TPB - 1) / TPB, TPB, 0, stream>>>(attn, 0.f, 3200000);
  exp_denom_kernel<<<edge_thr_blocks, TPB, 0, stream>>>(ei, mEnc, logitsp, den);
  attn_scatter_kernel<<<edge_wave_blocks, TPB, 0, stream>>>(ei, logitsp, den, vbuf, attn);

  // 5) residual + relu + classifier
  final_kernel<<<(N_NODES + TPB - 1) / TPB, TPB, 0, stream>>>(attn, skip, Wf, bfb, out);
}